// Model_81080392614473
// MI455X (gfx1250) — compile-verified
//
#include <hip/hip_runtime.h>
#include <cstdint>
#include <cstddef>

typedef __attribute__((ext_vector_type(2))) float v2f;
typedef __attribute__((ext_vector_type(8))) float v8f;

// ---------------------------------------------------------------------------
// Degree: deg[i] = 1 (self-loop) + #incoming edges; then dinv = rsqrt(deg).
// ---------------------------------------------------------------------------
__global__ void k_deg_init(float* __restrict__ deg, int n) {
  int i = blockIdx.x * blockDim.x + threadIdx.x;
  if (i < n) deg[i] = 1.0f;
}

__global__ void k_deg_accum(const long long* __restrict__ dst, float* __restrict__ deg,
                            long long e) {
  long long stride = (long long)gridDim.x * blockDim.x;
  for (long long i = (long long)blockIdx.x * blockDim.x + threadIdx.x; i < e; i += stride) {
    long long d = dst[i];
    __hip_atomic_fetch_add(&deg[d], 1.0f, __ATOMIC_RELAXED, __HIP_MEMORY_SCOPE_AGENT);
  }
}

__global__ void k_rsqrt_inplace(float* __restrict__ deg, int n) {
  int i = blockIdx.x * blockDim.x + threadIdx.x;
  if (i < n) deg[i] = rsqrtf(deg[i]);
}

// ---------------------------------------------------------------------------
// Dense GEMM H[N x M] = X[N x K] @ W[K x M] with V_WMMA_F32_16X16X4_F32.
// Each wave owns a 64x64 patch: RT=4 row tiles x MT=4 col tiles = 16
// accumulators. Fragments are double-buffered (ping-pong) so global loads for
// step k+4 are in flight while the 16 WMMAs for step k execute.
// Layouts per CDNA5 ISA 7.12.2:
//   A (16x4 f32): lane<16 -> K=0,1 ; lane>=16 -> K=2,3 ; M = lane&15
//   C/D (16x16 f32): VGPR r -> M = r + (lane>=16 ? 8 : 0); N = lane&15
// Tail handling: row-tile indices clamp to NT-1 (loads in-bounds, redundant
// compute), stores guarded by the wave-uniform (g+i < NT) test.
// ---------------------------------------------------------------------------
template <int K, int M>
__global__ __launch_bounds__(128) void k_gemm_wmma_f32(
    const float* __restrict__ X, const float* __restrict__ W, float* __restrict__ H,
    int NT /* row tiles = N/16 */) {
  constexpr int RT = 4;
  constexpr int MT = 4;
  const int wave = threadIdx.x >> 5;
  const int lane = threadIdx.x & 31;
  const int mrow = lane & 15;
  const int khalf = (lane >> 4) * 2;  // 0 or 2
  const int col0 = wave * (MT * 16);
  const int g = blockIdx.x * RT;

  int tile[RT];
  const float* arow[RT];
#pragma unroll
  for (int i = 0; i < RT; ++i) {
    int t = g + i;
    if (t > NT - 1) t = NT - 1;
    tile[i] = t;
    arow[i] = X + (size_t)(t * 16 + mrow) * K + khalf;
  }
  const float* wp[MT];
#pragma unroll
  for (int j = 0; j < MT; ++j)
    wp[j] = W + (size_t)khalf * M + (col0 + j * 16 + mrow);

  v8f acc[RT][MT];
#pragma unroll
  for (int i = 0; i < RT; ++i)
#pragma unroll
    for (int j = 0; j < MT; ++j) acc[i][j] = v8f{};

  v2f a0[RT], b0[MT], a1[RT], b1[MT];

  auto loadA = [&](int k, v2f* d) {
#pragma unroll
    for (int i = 0; i < RT; ++i) d[i] = *(const v2f*)(arow[i] + k);
  };
  auto loadB = [&](int k, v2f* d) {
#pragma unroll
    for (int j = 0; j < MT; ++j) {
      v2f b;
      b.x = wp[j][(size_t)k * M];
      b.y = wp[j][(size_t)k * M + M];
      d[j] = b;
    }
  };
  auto mma = [&](v2f* a, v2f* b) {
#pragma unroll
    for (int i = 0; i < RT; ++i)
#pragma unroll
      for (int j = 0; j < MT; ++j)
        acc[i][j] = __builtin_amdgcn_wmma_f32_16x16x4_f32(
            false, a[i], false, b[j], (short)0, acc[i][j], false, false);
  };

  loadA(0, a0);
  loadB(0, b0);
  for (int k = 0; k < K; k += 8) {          // K % 8 == 0
    loadA(k + 4, a1);                        // prefetch step k+4
    loadB(k + 4, b1);
    mma(a0, b0);                             // compute step k
    if (k + 8 < K) {                         // prefetch step k+8
      loadA(k + 8, a0);
      loadB(k + 8, b0);
    }
    mma(a1, b1);                             // compute step k+4
  }

  const int rhalf = (lane >> 4) * 8;
#pragma unroll
  for (int i = 0; i < RT; ++i) {
    if (g + i < NT) {
      int rowb = tile[i] * 16 + rhalf;
#pragma unroll
      for (int j = 0; j < MT; ++j) {
        int n = col0 + j * 16 + mrow;
#pragma unroll
        for (int r = 0; r < 8; ++r)
          H[(size_t)(rowb + r) * M + n] = acc[i][j][r];
      }
    }
  }
}

// ---------------------------------------------------------------------------
// out[i][c] = h[i][c] * dinv[i]^2 + bias[c]  (self-loop term + bias init),
// float4-wide. cqshift = log2(C/4).
// ---------------------------------------------------------------------------
__global__ void k_selfloop_bias4(const float4* __restrict__ H4, const float* __restrict__ dinv,
                                 const float4* __restrict__ bias4, float4* __restrict__ out4,
                                 long long total4, int cqshift) {
  long long stride = (long long)gridDim.x * blockDim.x;
  int cqmask = (1 << cqshift) - 1;
  for (long long i = (long long)blockIdx.x * blockDim.x + threadIdx.x; i < total4; i += stride) {
    int node = (int)(i >> cqshift);
    float di = dinv[node];
    float s = di * di;
    float4 h = H4[i];
    float4 b = bias4[i & cqmask];
    out4[i] = float4{h.x * s + b.x, h.y * s + b.y, h.z * s + b.z, h.w * s + b.w};
  }
}

// ---------------------------------------------------------------------------
// Edge scatter: out[dst] += h[src] * dinv[src]*dinv[dst]. One wave per edge
// (grid-stride); lanes read h[src] as float4 (global_load_b128) and issue
// no-return global_atomic_add_f32 per channel. h/out are L2-resident
// (51 MB << 192 MB L2), so this runs at the L2 f32-atomic rate.
// ---------------------------------------------------------------------------
__global__ void k_edge_scatter(const float* __restrict__ H, const long long* __restrict__ src,
                               const long long* __restrict__ dst, const float* __restrict__ dinv,
                               float* __restrict__ out, long long E, int Cq /* = C/4 */) {
  const int lane = threadIdx.x & 31;
  const long long nwaves = (long long)gridDim.x * (blockDim.x >> 5);
  long long w = (long long)blockIdx.x * (blockDim.x >> 5) + (threadIdx.x >> 5);
  for (long long e = w; e < E; e += nwaves) {
    long long s = src[e];
    long long d = dst[e];
    float nrm = dinv[s] * dinv[d];
    const float4* hs = (const float4*)(H + (size_t)s * (Cq * 4));
    float* od = out + (size_t)d * (Cq * 4);
    for (int q = lane; q < Cq; q += 32) {
      float4 v = hs[q];
      float* p = od + q * 4;
      __hip_atomic_fetch_add(p + 0, v.x * nrm, __ATOMIC_RELAXED, __HIP_MEMORY_SCOPE_AGENT);
      __hip_atomic_fetch_add(p + 1, v.y * nrm, __ATOMIC_RELAXED, __HIP_MEMORY_SCOPE_AGENT);
      __hip_atomic_fetch_add(p + 2, v.z * nrm, __ATOMIC_RELAXED, __HIP_MEMORY_SCOPE_AGENT);
      __hip_atomic_fetch_add(p + 3, v.w * nrm, __ATOMIC_RELAXED, __HIP_MEMORY_SCOPE_AGENT);
    }
  }
}

__global__ void k_relu4_inplace(float4* __restrict__ a, long long total4) {
  long long stride = (long long)gridDim.x * blockDim.x;
  for (long long i = (long long)blockIdx.x * blockDim.x + threadIdx.x; i < total4; i += stride) {
    float4 v = a[i];
    a[i] = float4{fmaxf(v.x, 0.f), fmaxf(v.y, 0.f), fmaxf(v.z, 0.f), fmaxf(v.w, 0.f)};
  }
}

// ---------------------------------------------------------------------------
// Orchestration
// ---------------------------------------------------------------------------
extern "C" void kernel_launch(void* const* d_in, const int* in_sizes, int n_in,
                              void* d_out, int out_size, void* d_ws, size_t ws_size,
                              hipStream_t stream) {
  constexpr int C_IN = 256, C_HID = 256, C_OUT = 128;
  const float*     x     = (const float*)d_in[0];
  const long long* edges = (const long long*)d_in[1];  // [2, E] int64
  const float*     W1    = (const float*)d_in[2];
  const float*     b1    = (const float*)d_in[3];
  const float*     W2    = (const float*)d_in[4];
  const float*     b2    = (const float*)d_in[5];
  float*           out   = (float*)d_out;

  const int       N  = in_sizes[0] / C_IN;          // 50000
  const int       NT = N / 16;                      // 3125 row tiles
  const long long E  = (long long)in_sizes[1] / 2;  // 1.6M
  const long long* esrc = edges;
  const long long* edst = edges + E;

  // Workspace layout
  char* ws = (char*)d_ws;
  size_t degBytes = (((size_t)N * 4) + 255) & ~(size_t)255;
  float* dinv = (float*)ws;               // N       (deg -> dinv, in place)
  float* h1   = (float*)(ws + degBytes);  // N*256
  float* a1   = h1 + (size_t)N * C_HID;   // N*256
  float* h2   = a1 + (size_t)N * C_HID;   // N*128
  (void)ws_size; (void)n_in; (void)out_size;

  const int TB = 256;
  dim3 b256(TB);

  // --- degrees -> dinv ---
  k_deg_init<<<dim3((N + TB - 1) / TB), b256, 0, stream>>>(dinv, N);
  k_deg_accum<<<dim3(2048), b256, 0, stream>>>(edst, dinv, E);
  k_rsqrt_inplace<<<dim3((N + TB - 1) / TB), b256, 0, stream>>>(dinv, N);

  // --- layer 1: h1 = x @ W1 ; a1 = self+bias ; scatter ; relu ---
  k_gemm_wmma_f32<256, 256><<<dim3((NT + 3) / 4), dim3(128), 0, stream>>>(x, W1, h1, NT);
  long long tot1q = (long long)N * C_HID / 4;
  k_selfloop_bias4<<<dim3(4096), b256, 0, stream>>>((const float4*)h1, dinv,
                                                    (const float4*)b1, (float4*)a1, tot1q, 6);
  k_edge_scatter<<<dim3(8192), b256, 0, stream>>>(h1, esrc, edst, dinv, a1, E, C_HID / 4);
  k_relu4_inplace<<<dim3(4096), b256, 0, stream>>>((float4*)a1, tot1q);

  // --- layer 2: h2 = a1 @ W2 ; d_out = self+bias ; scatter ; relu ---
  k_gemm_wmma_f32<256, 128><<<dim3((NT + 3) / 4), dim3(64), 0, stream>>>(a1, W2, h2, NT);
  long long tot2q = (long long)N * C_OUT / 4;
  k_selfloop_bias4<<<dim3(4096), b256, 0, stream>>>((const float4*)h2, dinv,
                                                    (const float4*)b2, (float4*)out, tot2q, 5);
  k_edge_scatter<<<dim3(8192), b256, 0, stream>>>(h2, esrc, edst, dinv, out, E, C_OUT / 4);
  k_relu4_inplace<<<dim3(4096), b256, 0, stream>>>((float4*)out, tot2q);
}